// EquivariantLinear_45938970198757
// MI455X (gfx1250) — compile-verified
//
#include <hip/hip_runtime.h>

// ---------------------------------------------------------------------------
// EquivariantLinear (irreps 0e+1o+2e, mix_channels) for MI455X / gfx1250.
//   out[bn, c, m] = (1/16) * sum_k x[bn, k, m] * W_{l(m)}[k, c]
// B=8, N=4096 -> 32768 sites; C_in=C_out=256; FEAT=9 (dims 1,3,5).
// fp32 compute-bound => V_WMMA_F32_16X16X4_F32.
// All 9 m-tiles accumulated simultaneously (72 acc VGPRs, 9 independent WMMA
// chains); B fragments (6 floats / K-step) come from global (WGP$/L2-hot);
// x staged per-site by TDM into LDS with a 4-dword inter-site gap
// (stride 2308 % 64 == 4 -> conflict-free A reads), double buffered.
// K-loop unrolled x8 to stay under 256 live VGPRs (no s_set_vgpr_msb churn).
// ---------------------------------------------------------------------------

typedef __attribute__((ext_vector_type(2)))  float        v2f;
typedef __attribute__((ext_vector_type(8)))  float        v8f;
typedef __attribute__((ext_vector_type(4)))  unsigned int v4u;
typedef __attribute__((ext_vector_type(8)))  int          v8i;
typedef __attribute__((ext_vector_type(4)))  int          v4i;

#define SITES       32768
#define CIN         256
#define COUT        256
#define FEAT        9
#define SPG         16                    // sites per group = WMMA M dim
#define SITE_DW     (CIN * FEAT)          // 2304 dwords per site
#define SITE_DW_PAD (SITE_DW + 4)         // 2308: stride%64==4 -> 16 banks
#define BUF_DW      (SPG * SITE_DW_PAD)   // 36928 dwords per LDS buffer
#define NGRP_BLK    8                     // groups per block (pipelined)
#define PW          0.0625f               // path weight 1/sqrt(256)

#if defined(__HIP_DEVICE_COMPILE__)
# if !__has_builtin(__builtin_amdgcn_wmma_f32_16x16x4_f32)
#  error "gfx1250 f32 WMMA builtin __builtin_amdgcn_wmma_f32_16x16x4_f32 not available"
# endif
# if __has_builtin(__builtin_amdgcn_tensor_load_to_lds) && \
     __has_builtin(__builtin_amdgcn_s_wait_tensorcnt)
#  define USE_TDM 1
# endif
#endif

__device__ static inline v8f wmma_f32_16x16x4(v2f a, v2f b, v8f c)
{
#if defined(__HIP_DEVICE_COMPILE__)
    return __builtin_amdgcn_wmma_f32_16x16x4_f32(false, a, false, b,
                                                 (short)0, c, false, false);
#else
    return c;  // host parse stub
#endif
}

// 1D TDM copy: one contiguous site block (2304 dwords) global -> LDS.
__device__ static inline void tdm_site(const float* gsrc, unsigned ldsByteOff)
{
#ifdef USE_TDM
    unsigned long long ga = (unsigned long long)gsrc;
    v4u g0;
    g0.x = 1u;                                              // count=1 user D#
    g0.y = ldsByteOff;                                      // lds_addr
    g0.z = (unsigned)ga;                                    // global_addr[31:0]
    g0.w = (unsigned)((ga >> 32) & 0x01FFFFFFu) | (2u << 30); // [56:32]|type=2
    v8i g1;
    g1[0] = 0x00020000;           // data_size=4B, no pad, no multicast
    g1[1] = (int)(2304u << 16);   // tensor_dim0 = 2304 (low16)
    g1[2] = (int)(1u << 16);      // tensor_dim0 hi=0 | tensor_dim1 = 1
    g1[3] = (int)(2304u << 16);   // tensor_dim1 hi=0 | tile_dim0 = 2304
    g1[4] = 0;                    // tile_dim1 = 0 (unused), tile_dim2 = 0
    g1[5] = 2304;                 // tensor_dim0_stride
    g1[6] = 0;
    g1[7] = 0;
    v4i z4 = {0, 0, 0, 0};
    v8i z8 = {0, 0, 0, 0, 0, 0, 0, 0};
    __builtin_amdgcn_tensor_load_to_lds(g0, g1, z4, z4, z8, 0);
#else
    (void)gsrc; (void)ldsByteOff;
#endif
}

// Stage one group (16 sites) with per-site LDS gap of 4 dwords.
__device__ static inline void stage_group(const float* xg, unsigned bufByte)
{
#pragma unroll
    for (int s = 0; s < SPG; ++s)
        tdm_site(xg + s * SITE_DW, bufByte + (unsigned)s * (SITE_DW_PAD * 4));
}

__global__ __launch_bounds__(256)
void equi_linear_wmma(const float* __restrict__ x,
                      const float* __restrict__ w0,
                      const float* __restrict__ w1,
                      const float* __restrict__ w2,
                      float* __restrict__ out)
{
    extern __shared__ float smem[];

    const int tid   = threadIdx.x;
    const int wave  = tid >> 5;
    const int lane  = tid & 31;
    const int half  = lane >> 4;      // 0: lanes 0-15, 1: lanes 16-31
    const int ln16  = lane & 15;
    const int kOff  = half * 2;       // K sub-offset per ISA A/B VGPR layout
    const int cLane = blockIdx.y * 128 + wave * 16 + ln16;
    const int grpBase = blockIdx.x * NGRP_BLK;

    // Per-irrep weight row pointers: element k=4t+kOff is wr[l][t*1024],
    // k+1 is wr[l][t*1024 + 256]. 16 consecutive lanes -> 64B segments.
    const float* wr[3] = { w0 + kOff * COUT + cLane,
                           w1 + kOff * COUT + cLane,
                           w2 + kOff * COUT + cLane };

#ifdef USE_TDM
    if (wave == 0)
        stage_group(x + (long)grpBase * SPG * SITE_DW, 0u);
#endif

#pragma unroll 1
    for (int g = 0; g < NGRP_BLK; ++g) {
#ifdef USE_TDM
        if (wave == 0) {
            if (g + 1 < NGRP_BLK) {
                stage_group(x + (long)(grpBase + g + 1) * SPG * SITE_DW,
                            (unsigned)(((g + 1) & 1) * BUF_DW * 4));
                __builtin_amdgcn_s_wait_tensorcnt(16); // group g's 16 done
            } else {
                __builtin_amdgcn_s_wait_tensorcnt(0);
            }
        }
        __syncthreads();
#else
        __syncthreads();
        {   // fallback cooperative staging with the same gapped layout
            const float* gsrc = x + (long)(grpBase + g) * SPG * SITE_DW;
            const int bufDwF = (g & 1) * BUF_DW;
            for (int s = 0; s < SPG; ++s)
                for (int i = tid; i < SITE_DW; i += 256)
                    smem[bufDwF + s * SITE_DW_PAD + i] = gsrc[s * SITE_DW + i];
        }
        __syncthreads();
#endif

        // ---- compute: 9 simultaneous 16x16 tiles, K = 256 in 64 steps ----
        const int aBase = (g & 1) * BUF_DW + ln16 * SITE_DW_PAD + kOff * FEAT;

        v8f acc[FEAT];
#pragma unroll
        for (int m = 0; m < FEAT; ++m)
            acc[m] = (v8f){0.f, 0.f, 0.f, 0.f, 0.f, 0.f, 0.f, 0.f};

#pragma unroll 8
        for (int t = 0; t < 64; ++t) {
            // B fragments: one per irrep, shared across that irrep's m-tiles
            v2f b[3];
#pragma unroll
            for (int l = 0; l < 3; ++l) {
                b[l].x = wr[l][t * 1024];        // W[4t+kOff  , cLane]
                b[l].y = wr[l][t * 1024 + 256];  // W[4t+kOff+1, cLane]
            }
            const int aIdx = aBase + 36 * t;     // x[.,4t+kOff,0] dword index
#pragma unroll
            for (int m = 0; m < FEAT; ++m) {
                const int l = (m == 0) ? 0 : (m < 4) ? 1 : 2;
                v2f a;
                a.x = smem[aIdx + m];            // x[k  , m]
                a.y = smem[aIdx + m + FEAT];     // x[k+1, m]
                acc[m] = wmma_f32_16x16x4(a, b[l], acc[m]);
            }
        }

        // ---- store: D vgpr r -> M = half*8 + r (site), N = ln16 (channel)
        const int siteBase = (grpBase + g) * SPG;
#pragma unroll
        for (int m = 0; m < FEAT; ++m) {
#pragma unroll
            for (int r = 0; r < 8; ++r) {
                const long site = siteBase + half * 8 + r;
                float* dst = out + (site * COUT + cLane) * FEAT + m;
                __builtin_nontemporal_store(acc[m][r] * PW, dst);
            }
        }
        __syncthreads();   // buffer (g&1) free for TDM reuse two iterations on
    }
}

extern "C" void kernel_launch(void* const* d_in, const int* in_sizes, int n_in,
                              void* d_out, int out_size, void* d_ws, size_t ws_size,
                              hipStream_t stream)
{
    const float* x  = (const float*)d_in[0];
    const float* w0 = (const float*)d_in[1];
    const float* w1 = (const float*)d_in[2];
    const float* w2 = (const float*)d_in[3];
    float* out = (float*)d_out;

    dim3 grid(SITES / (SPG * NGRP_BLK), 2, 1);     // 256 x 2 blocks
    dim3 block(256, 1, 1);                          // 8 waves (wave32)
    size_t shmem = (size_t)2 * BUF_DW * sizeof(float);  // 295424 B LDS
    hipLaunchKernelGGL(equi_linear_wmma, grid, block, shmem, stream,
                       x, w0, w1, w2, out);
}